// TemporalGNN_32658931319165
// MI455X (gfx1250) — compile-verified
//
#include <hip/hip_runtime.h>
#include <math.h>

typedef __attribute__((ext_vector_type(2))) float v2f;
typedef __attribute__((ext_vector_type(8))) float v8f;

#define N_NODES 16384
#define IN_DIM  4
#define HIDDEN  64
#define OUT_DIM 2
#define KC      64                 // K-chunk staged in LDS
#define NCHUNK  (N_NODES / KC)     // 256
#define PRS     160                // dwords per LDS pair-row (32 pair-rows/chunk)

// ---------------------------------------------------------------------------
// Stage 1: messages = tanh(x @ W1 + b1) @ W2 + b2        (16384 x 64)
// ---------------------------------------------------------------------------
__global__ __launch_bounds__(256) void gnn_msg_kernel(
    const float* __restrict__ x, const float* __restrict__ W1,
    const float* __restrict__ b1, const float* __restrict__ W2,
    const float* __restrict__ b2, float* __restrict__ msg) {
  __shared__ float h[4][HIDDEN];
  const int tid = threadIdx.x;
  const int r = tid >> 6;
  const int j = tid & 63;
  const int row = blockIdx.x * 4 + r;

  const float* xr = x + (size_t)row * IN_DIM;
  float s = b1[j];
#pragma unroll
  for (int d = 0; d < IN_DIM; ++d) s += xr[d] * W1[d * HIDDEN + j];
  h[r][j] = tanhf(s);
  __syncthreads();

  float m = b2[j];
#pragma unroll 8
  for (int k = 0; k < HIDDEN; ++k) m += h[r][k] * W2[k * HIDDEN + j];
  msg[(size_t)row * HIDDEN + j] = m;
}

// ---------------------------------------------------------------------------
// Stage 2: agg = adj @ messages, fp32 WMMA.
// LDS B layout: element (k, n) at float index (k>>1)*PRS + 2*n + (k&1),
// so a lane's (k, k+1) B-fragment is one aligned ds_load_b64.
// ---------------------------------------------------------------------------
__device__ __forceinline__ v8f wmma4(const v2f a, const v2f b, const v8f c) {
  return __builtin_amdgcn_wmma_f32_16x16x4_f32(false, a, false, b, (short)0, c,
                                               false, false);
}

// bsrc: per-thread msg base (msg + 2*wave*HIDDEN + 2*cp), offset by k0n*HIDDEN
// ldsBase: per-thread LDS store base (buf + wave*PRS + 4*cp)
__device__ __forceinline__ void agg_chunk(
    const int c, const float* __restrict__ arow, const int koff, const int m,
    const float* __restrict__ bsrc, const float* __restrict__ bsCur,
    float* __restrict__ ldsNxt,
    v2f (&aCur)[16], v2f (&aNxt)[16],
    v8f& acc0, v8f& acc1, v8f& acc2, v8f& acc3) {
  const int k0 = c * KC;
  const bool hasNext = (c + 1 < NCHUNK);

  // HBM prefetch 3 chunks ahead into all cache levels (locality 3 -> near
  // scope); half-wave pair covers the full 256B of each of the 16 rows.
  if (c + 3 < NCHUNK) {
    __builtin_prefetch(arow + k0 + 3 * KC + ((koff >> 1) << 5), 0, 3);
  }

  // Issue next chunk's A fragments (16 outstanding b64 NT loads) and B tile
  // global loads BEFORE this chunk's compute, so they overlap the WMMAs.
  float4 breg[8];
  if (hasNext) {
    const int k0n = k0 + KC;
#pragma unroll
    for (int i = 0; i < 16; ++i) {
      aNxt[i] = __builtin_nontemporal_load(
          (const v2f*)(arow + k0n + i * 4 + koff));
    }
    // all 16 loads are immediate offsets from one base pointer
    const float* src = bsrc + (size_t)k0n * HIDDEN;
#pragma unroll
    for (int i = 0; i < 8; ++i) {
      const v2f lo = *(const v2f*)(src + i * 8 * HIDDEN);           // row 8i+2w
      const v2f hi = *(const v2f*)(src + i * 8 * HIDDEN + HIDDEN);  // row +1
      breg[i] = make_float4(lo.x, hi.x, lo.y, hi.y);
    }
  }

  // ---- 16 k-steps of 4: 64 x V_WMMA_F32_16X16X4_F32 ----
#pragma unroll
  for (int kl = 0; kl < KC; kl += 4) {
    const v2f a = aCur[kl >> 2];
    const float* bp = bsCur + ((kl >> 1) + (koff >> 1)) * PRS + 2 * m;
    const v2f b0 = *(const v2f*)(bp);
    const v2f b1 = *(const v2f*)(bp + 32);
    const v2f b2 = *(const v2f*)(bp + 64);
    const v2f b3 = *(const v2f*)(bp + 96);
    acc0 = wmma4(a, b0, acc0);
    acc1 = wmma4(a, b1, acc1);
    acc2 = wmma4(a, b2, acc2);
    acc3 = wmma4(a, b3, acc3);
  }

  // Commit next B tile (pair-interleaved layout), immediate offsets only.
  if (hasNext) {
#pragma unroll
    for (int i = 0; i < 8; ++i) {
      *(float4*)(ldsNxt + i * 4 * PRS) = breg[i];
    }
  }
  __syncthreads();
}

__global__ __launch_bounds__(128) void gnn_agg_wmma(
    const float* __restrict__ adj, const float* __restrict__ msg,
    float* __restrict__ agg) {
  __shared__ float Bs[2][32 * PRS];   // 2 x 20 KB

  const int tid  = threadIdx.x;
  const int wave = tid >> 5;
  const int lane = tid & 31;
  const int m    = lane & 15;          // A-matrix row within tile
  const int koff = (lane >> 4) << 1;   // 0 (lanes 0-15) / 2 (lanes 16-31)
  const int cp   = tid & 31;           // column-pair for B staging
  const int rowBase = blockIdx.x * 64 + wave * 16;

  const float* arow = adj + (size_t)(rowBase + m) * N_NODES;
  const float* bsrc = msg + (size_t)(2 * wave) * HIDDEN + 2 * cp;
  float* lds0 = &Bs[0][0] + wave * PRS + 4 * cp;
  float* lds1 = &Bs[1][0] + wave * PRS + 4 * cp;

  v8f acc0 = {}, acc1 = {}, acc2 = {}, acc3 = {};
  v2f areg0[16], areg1[16];

  // ---- prologue: A fragments for chunk 0, B chunk 0 -> Bs[0] ----
#pragma unroll
  for (int i = 0; i < 16; ++i) {
    areg0[i] = __builtin_nontemporal_load((const v2f*)(arow + i * 4 + koff));
  }
#pragma unroll
  for (int i = 0; i < 8; ++i) {
    const v2f lo = *(const v2f*)(bsrc + i * 8 * HIDDEN);
    const v2f hi = *(const v2f*)(bsrc + i * 8 * HIDDEN + HIDDEN);
    *(float4*)(lds0 + i * 4 * PRS) = make_float4(lo.x, hi.x, lo.y, hi.y);
  }
  __syncthreads();

  // ---- main loop, unrolled by 2 so both double-buffers resolve statically --
  for (int c = 0; c < NCHUNK; c += 2) {
    agg_chunk(c,     arow, koff, m, bsrc, &Bs[0][0], lds1,
              areg0, areg1, acc0, acc1, acc2, acc3);
    agg_chunk(c + 1, arow, koff, m, bsrc, &Bs[1][0], lds0,
              areg1, areg0, acc0, acc1, acc2, acc3);
  }

  // ---- write C: VGPR r holds (M=r, lanes 0-15) / (M=8+r, lanes 16-31) ----
  const int mhi = (lane >> 4) * 8;
  const int nlo = lane & 15;
#pragma unroll
  for (int r = 0; r < 8; ++r) {
    float* o = agg + (size_t)(rowBase + mhi + r) * HIDDEN + nlo;
    o[0]  = acc0[r];
    o[16] = acc1[r];
    o[32] = acc2[r];
    o[48] = acc3[r];
  }
}

// ---------------------------------------------------------------------------
// Stage 3: h = tanh(tanh([x, agg] @ U1 + c1) @ U2 + c2); out = h @ Wo + bo
// ---------------------------------------------------------------------------
__global__ __launch_bounds__(256) void gnn_update_kernel(
    const float* __restrict__ x, const float* __restrict__ agg,
    const float* __restrict__ U1, const float* __restrict__ c1,
    const float* __restrict__ U2, const float* __restrict__ c2,
    const float* __restrict__ Wo, const float* __restrict__ bo,
    float* __restrict__ out) {
  __shared__ float ag[4][HIDDEN];
  __shared__ float h1[4][HIDDEN];
  __shared__ float h2[4][HIDDEN];
  const int tid = threadIdx.x;
  const int r = tid >> 6;
  const int j = tid & 63;
  const int row = blockIdx.x * 4 + r;

  ag[r][j] = agg[(size_t)row * HIDDEN + j];
  __syncthreads();

  const float* xr = x + (size_t)row * IN_DIM;
  float s = c1[j];
#pragma unroll
  for (int d = 0; d < IN_DIM; ++d) s += xr[d] * U1[d * HIDDEN + j];
#pragma unroll 8
  for (int k = 0; k < HIDDEN; ++k)
    s += ag[r][k] * U1[(IN_DIM + k) * HIDDEN + j];
  h1[r][j] = tanhf(s);
  __syncthreads();

  float t = c2[j];
#pragma unroll 8
  for (int k = 0; k < HIDDEN; ++k) t += h1[r][k] * U2[k * HIDDEN + j];
  h2[r][j] = tanhf(t);
  __syncthreads();

  if (j < OUT_DIM) {
    float o = bo[j];
#pragma unroll 8
    for (int k = 0; k < HIDDEN; ++k) o += h2[r][k] * Wo[k * OUT_DIM + j];
    out[(size_t)row * OUT_DIM + j] = o;
  }
}

// ---------------------------------------------------------------------------
extern "C" void kernel_launch(void* const* d_in, const int* in_sizes, int n_in,
                              void* d_out, int out_size, void* d_ws,
                              size_t ws_size, hipStream_t stream) {
  const float* x   = (const float*)d_in[0];
  const float* adj = (const float*)d_in[1];
  const float* W1  = (const float*)d_in[2];
  const float* b1  = (const float*)d_in[3];
  const float* W2  = (const float*)d_in[4];
  const float* b2  = (const float*)d_in[5];
  const float* U1  = (const float*)d_in[6];
  const float* c1  = (const float*)d_in[7];
  const float* U2  = (const float*)d_in[8];
  const float* c2  = (const float*)d_in[9];
  const float* Wo  = (const float*)d_in[10];
  const float* bo  = (const float*)d_in[11];
  float* out = (float*)d_out;

  float* msg = (float*)d_ws;                       // 16384*64 floats (4 MB)
  float* agg = msg + (size_t)N_NODES * HIDDEN;     // 16384*64 floats (4 MB)

  gnn_msg_kernel<<<N_NODES / 4, 256, 0, stream>>>(x, W1, b1, W2, b2, msg);
  gnn_agg_wmma<<<N_NODES / 64, 128, 0, stream>>>(adj, msg, agg);
  gnn_update_kernel<<<N_NODES / 4, 256, 0, stream>>>(x, agg, U1, c1, U2, c2,
                                                     Wo, bo, out);
}